// symLoss_35957466202452
// MI455X (gfx1250) — compile-verified
//
#include <hip/hip_runtime.h>
#include <math.h>

// ---------------------------------------------------------------------------
// symLoss on MI455X (gfx1250, wave32)
//
// Strategy:
//  * One V_WMMA_F32_16X16X4_F32 per 16-point tile computes BOTH the plane
//    reflection and the quaternion rotation for TWO k-indices at once:
//      A (16x4): rows 0-2 refl(k_lo) 3x4 affine, rows 3-5 rot(k_lo) 3x4,
//                rows 8-10 refl(k_hi), rows 11-13 rot(k_hi), rest zero.
//      B (4x16): 16 points as columns (x,y,z,1).
//      D (16x16): lane l (col = l%16, half h = l/16) receives rows r+8h in
//                 d[r]  ->  refl xyz = d[0..2], rot xyz = d[3..5] for its k.
//  * Each lane then does the voxel-grid index + cp gather (single
//    global_load_b96 w/ scale_offset) + squared distance for its point/k,
//    for both transform types.  K=3 is odd -> blockIdx.z selects k-pair
//    {0,1} or {2,dup}; the duplicated half is predicated off (EXEC-masked
//    gathers, no wasted traffic).  WMMA always sees full EXEC because the
//    tile loop is wave-uniform and the B-operand load is branchless.
//  * Deterministic 2-stage reduction: per-block partials in d_ws, then one
//    fixed-order reduction block writes the 2 output scalars.
//
// VGPR layout assumptions (CDNA5 ISA):
//   A 16x4 f32: lanes 0-15 hold {K0,K1}, lanes 16-31 hold {K2,K3}, M = l%16.
//   B 4x16 f32: assumed mirrored split: vgpr0 = rows K0|K2, vgpr1 = K1|K3,
//               N = l%16 (row striped across lanes, halves offset by 2 rows).
//   D 16x16 f32: vgpr r: lanes 0-15 -> M=r, lanes 16-31 -> M=r+8, N = l%16.
// ---------------------------------------------------------------------------

namespace {
constexpr int GS  = 64;
constexpr int GS3 = GS * GS * GS;
constexpr int NB  = 128;   // batches
constexpr int NP  = 8192;  // points per batch
}

typedef __attribute__((ext_vector_type(2))) float v2f;
typedef __attribute__((ext_vector_type(8))) float v8f;

__device__ __forceinline__ float gather_dist(const float* __restrict__ cpB,
                                             float x, float y, float z) {
  // idx = clip(floor((p/(2*0.5)+0.5)*64), 0, 63)  (clip in float, then cast,
  // matching jnp order; clamp-before-cast also guards huge reflections)
  float fx = floorf((x + 0.5f) * 64.0f);
  float fy = floorf((y + 0.5f) * 64.0f);
  float fz = floorf((z + 0.5f) * 64.0f);
  fx = fminf(fmaxf(fx, 0.0f), 63.0f);
  fy = fminf(fmaxf(fy, 0.0f), 63.0f);
  fz = fminf(fmaxf(fz, 0.0f), 63.0f);
  int flat = ((int)fx * GS + (int)fy) * GS + (int)fz;
  const float* nb = cpB + (size_t)flat * 3;   // -> global_load_b96 scale_offset
  float dx = x - nb[0], dy = y - nb[1], dz = z - nb[2];
  return dx * dx + dy * dy + dz * dz;
}

__global__ __launch_bounds__(256) void sym_stage1(
    const float* __restrict__ points,  // (B, N, 3)
    const float* __restrict__ cp,      // (B, GS^3, 3)
    const float* __restrict__ plane,   // (K, B, 4)
    const float* __restrict__ quat,    // (K, B, 4)
    float* __restrict__ partial)       // (numBlocks, 2)
{
  const int lane = threadIdx.x & 31;
  const int wave = threadIdx.x >> 5;
  const int col  = lane & 15;   // = N column of B/D, = M row of A
  const int half = lane >> 4;
  const int b    = blockIdx.y;
  const int kpair = blockIdx.z;           // 0 -> ks {0,1}, 1 -> ks {2, dup}
  const int k_lo = kpair * 2;             // D rows 0-7
  const int k_hi = (kpair == 0) ? 1 : 2;  // D rows 8-15 (dup of k=2 if kpair=1)
  const bool live = !(kpair == 1 && half == 1);  // mask duplicated half

  // ---- Build loop-invariant A operand: lane holds A[col][2h], A[col][2h+1]
  const int lm = col & 7;                    // 0-2 refl row, 3-5 rot row, 6-7 zero
  const int kA = (col < 8) ? k_lo : k_hi;    // which transform this A-row belongs to
  const float* pl = plane + (size_t)(kA * NB + b) * 4;
  const float* qq = quat  + (size_t)(kA * NB + b) * 4;

  // reflection: p' = (I - (2/s) n n^T) p - (2/s) d n,  s = |n|^2 + 1e-12
  const float nx = pl[0], ny = pl[1], nz = pl[2], pd = pl[3];
  const float s    = nx * nx + ny * ny + nz * nz + 1e-12f;
  const float inv2 = 2.0f / s;

  // rotation: q /= (|q| + 1e-12); p' = p + 2 v x (v x p + w p)
  //           => R = (1 - 2 v.v) I + 2 v v^T + 2 w [v]_x
  float qw = qq[0], qx = qq[1], qy = qq[2], qz = qq[3];
  const float qrn = 1.0f / (sqrtf(qw * qw + qx * qx + qy * qy + qz * qz) + 1e-12f);
  qw *= qrn; qx *= qrn; qy *= qrn; qz *= qrn;
  const float diag = 1.0f - 2.0f * (qx * qx + qy * qy + qz * qz);

  auto aelem = [&](int kc) -> float {
    if (lm < 3) {                      // reflect rows
      float ni = (lm == 0) ? nx : ((lm == 1) ? ny : nz);
      if (kc == 3) return -inv2 * pd * ni;        // translation column
      float nj = (kc == 0) ? nx : ((kc == 1) ? ny : nz);
      return ((kc == lm) ? 1.0f : 0.0f) - inv2 * ni * nj;
    } else if (lm < 6) {               // rotate rows (no translation)
      if (kc == 3) return 0.0f;
      int i = lm - 3, j = kc;
      float vi = (i == 0) ? qx : ((i == 1) ? qy : qz);
      float vj = (j == 0) ? qx : ((j == 1) ? qy : qz);
      float r = 2.0f * vi * vj + ((i == j) ? diag : 0.0f);
      float cr;                        // [v]_x[i][j]
      if (i == 0)      cr = (j == 1) ? -qz : ((j == 2) ?  qy : 0.0f);
      else if (i == 1) cr = (j == 0) ?  qz : ((j == 2) ? -qx : 0.0f);
      else             cr = (j == 0) ? -qy : ((j == 1) ?  qx : 0.0f);
      return r + 2.0f * qw * cr;
    }
    return 0.0f;                       // rows 6,7 / 14,15
  };
  v2f amat;
  amat.x = aelem(2 * half);
  amat.y = aelem(2 * half + 1);

  const float* ptsB = points + (size_t)b * NP * 3;
  const float* cpB  = cp + (size_t)b * GS3 * 3;

  const int tilesTotal    = NP / 16;           // 512
  const int wavesPerBlock = blockDim.x >> 5;   // 8
  const int waveStride    = gridDim.x * wavesPerBlock;

  float acc_r = 0.0f, acc_o = 0.0f;
  for (int tile = blockIdx.x * wavesPerBlock + wave; tile < tilesTotal;
       tile += waveStride) {
    // ---- B operand: column col = point tile*16+col, (x,y,z,1) down rows.
    // Branchless: every lane loads the full point (one b96), then selects;
    // both half-waves hit the same two cachelines, no EXEC gymnastics.
    const float* pp = ptsB + (size_t)(tile * 16 + col) * 3;
    const float px = pp[0], py = pp[1], pz = pp[2];
    v2f bmat;
    bmat.x = half ? pz : px;    // rows K0 | K2
    bmat.y = half ? 1.0f : py;  // rows K1 | K3

    v8f c = {0.f, 0.f, 0.f, 0.f, 0.f, 0.f, 0.f, 0.f};
    // D = A(16x4) * B(4x16) + 0 : transforms 16 points by 4 affine matrices
    v8f d = __builtin_amdgcn_wmma_f32_16x16x4_f32(
        /*neg_a=*/false, amat, /*neg_b=*/false, bmat,
        /*c_mod=*/(short)0, c, /*reuse_a=*/false, /*reuse_b=*/false);

    if (live) {  // EXEC-masked gathers; WMMA above ran with full EXEC
      acc_r += gather_dist(cpB, d[0], d[1], d[2]);  // reflected point
      acc_o += gather_dist(cpB, d[3], d[4], d[5]);  // rotated point
    }
  }

  // ---- block reduction -> per-block partial (deterministic within block)
  __shared__ float sr[256];
  __shared__ float so[256];
  sr[threadIdx.x] = acc_r;
  so[threadIdx.x] = acc_o;
  __syncthreads();
  for (int off = 128; off > 0; off >>= 1) {
    if (threadIdx.x < (unsigned)off) {
      sr[threadIdx.x] += sr[threadIdx.x + off];
      so[threadIdx.x] += so[threadIdx.x + off];
    }
    __syncthreads();
  }
  if (threadIdx.x == 0) {
    int blin = blockIdx.x + gridDim.x * (blockIdx.y + gridDim.y * blockIdx.z);
    partial[2 * blin + 0] = sr[0];
    partial[2 * blin + 1] = so[0];
  }
}

__global__ __launch_bounds__(256) void sym_stage2(
    const float* __restrict__ partial, int nblocks, float* __restrict__ out) {
  __shared__ float sr[256];
  __shared__ float so[256];
  float ar = 0.0f, ao = 0.0f;
  for (int i = threadIdx.x; i < nblocks; i += 256) {
    ar += partial[2 * i + 0];
    ao += partial[2 * i + 1];
  }
  sr[threadIdx.x] = ar;
  so[threadIdx.x] = ao;
  __syncthreads();
  for (int off = 128; off > 0; off >>= 1) {
    if (threadIdx.x < (unsigned)off) {
      sr[threadIdx.x] += sr[threadIdx.x + off];
      so[threadIdx.x] += so[threadIdx.x + off];
    }
    __syncthreads();
  }
  if (threadIdx.x == 0) {
    const float scale = 1.0f / ((float)NB * (float)NP);  // sum_k of mean_{B,N}
    out[0] = sr[0] * scale;  // ref_loss
    out[1] = so[0] * scale;  // rot_loss
  }
}

extern "C" void kernel_launch(void* const* d_in, const int* in_sizes, int n_in,
                              void* d_out, int out_size, void* d_ws, size_t ws_size,
                              hipStream_t stream) {
  (void)in_sizes; (void)n_in; (void)out_size;
  const float* points = (const float*)d_in[0];  // (128, 8192, 3)
  const float* cp     = (const float*)d_in[1];  // (128, 64^3, 3)
  // d_in[2] = voxel, unused by reference
  const float* plane  = (const float*)d_in[3];  // (3, 128, 4)
  const float* quat   = (const float*)d_in[4];  // (3, 128, 4)
  float* out = (float*)d_out;
  float* ws  = (float*)d_ws;

  // pick gridDim.x so partials fit in d_ws (deterministic per environment)
  int gx;
  if      (ws_size >= (size_t)16 * NB * 2 * 2 * sizeof(float)) gx = 16;
  else if (ws_size >= (size_t)4  * NB * 2 * 2 * sizeof(float)) gx = 4;
  else                                                         gx = 1;
  const int nblocks = gx * NB * 2;

  dim3 grid(gx, NB, 2), block(256);
  sym_stage1<<<grid, block, 0, stream>>>(points, cp, plane, quat, ws);
  sym_stage2<<<1, 256, 0, stream>>>(ws, nblocks, out);
}